// LearnedChannelSelfAttentionPooling_V3_38757784879240
// MI455X (gfx1250) — compile-verified
//
#include <hip/hip_runtime.h>

// ---------------- problem constants ----------------
#define TDIM 512          // teacher channels
#define CS   8            // chunk size
#define NC   64           // num chunks
#define NB   8            // batch
#define HWP  9216         // 96*96 pixels
#define NPAIR (NB * NC)   // 512 (b,n) pairs
#define PRM  44           // 36 (upper-tri 8x8) + 8 (vector)
#define SPLIT3 9          // pixel splits for pass3 (9216 = 9*1024)

typedef __attribute__((ext_vector_type(2))) float v2f;
typedef __attribute__((ext_vector_type(8))) float v8f;

// D(16x16,f32) = A(16x4,f32) * B(4x16,f32) + C  — CDNA5 WMMA, wave32
__device__ __forceinline__ v8f wmma4(v2f a, v2f b, v8f c) {
  return __builtin_amdgcn_wmma_f32_16x16x4_f32(false, a, false, b, (short)0, c,
                                               false, false);
}

// symmetric 8x8 stored as upper triangle of 36
__device__ __forceinline__ float getG(const float* Gt, int i, int j) {
  if (i > j) { int t = i; i = j; j = t; }
  return Gt[8 * i - ((i * (i - 1)) >> 1) + (j - i)];
}

// ---------------- pass 1: per-pair Gram matrix G = xc*xc^T and s = xc*1 ----
// one block (256 threads = 8 waves) per (b,n) pair; streams 8ch x 9216px
__global__ void __launch_bounds__(256) lcsap_pass1(const float* __restrict__ x,
                                                   float* __restrict__ wsG) {
  __shared__ float red[8 * PRM];
  const int pair = blockIdx.x;
  const int b = pair >> 6, n = pair & 63;
  const float* xb = x + (size_t)(b * TDIM + n * CS) * HWP;

  float g[36], s[8];
#pragma unroll
  for (int k = 0; k < 36; ++k) g[k] = 0.f;
#pragma unroll
  for (int k = 0; k < 8; ++k) s[k] = 0.f;

  const int p0 = threadIdx.x * 4;
  for (int it = 0; it < 9; ++it) {         // 9 * 256 threads * 4 px = 9216
    const int p = p0 + (it << 10);
    float v[8][4];
#pragma unroll
    for (int c = 0; c < 8; ++c) {
      float4 t4 = *reinterpret_cast<const float4*>(xb + c * HWP + p);
      v[c][0] = t4.x; v[c][1] = t4.y; v[c][2] = t4.z; v[c][3] = t4.w;
    }
#pragma unroll
    for (int px = 0; px < 4; ++px) {
      int kidx = 0;
#pragma unroll
      for (int i = 0; i < 8; ++i) {
        s[i] += v[i][px];
#pragma unroll
        for (int j = i; j < 8; ++j) {
          g[kidx] = fmaf(v[i][px], v[j][px], g[kidx]);
          ++kidx;
        }
      }
    }
  }
  // wave-level xor-tree reduction (wave32)
#pragma unroll
  for (int k = 0; k < 36; ++k)
#pragma unroll
    for (int m = 16; m >= 1; m >>= 1) g[k] += __shfl_xor(g[k], m, 32);
#pragma unroll
  for (int k = 0; k < 8; ++k)
#pragma unroll
    for (int m = 16; m >= 1; m >>= 1) s[k] += __shfl_xor(s[k], m, 32);

  const int lane = threadIdx.x & 31, lw = threadIdx.x >> 5;
  if (lane == 0) {
#pragma unroll
    for (int k = 0; k < 36; ++k) red[lw * PRM + k] = g[k];
#pragma unroll
    for (int k = 0; k < 8; ++k) red[lw * PRM + 36 + k] = s[k];
  }
  __syncthreads();
  if (threadIdx.x < PRM) {
    float acc = 0.f;
#pragma unroll
    for (int w = 0; w < 8; ++w) acc += red[w * PRM + threadIdx.x];
    wsG[(size_t)pair * PRM + threadIdx.x] = acc;   // deterministic, no atomics
  }
}

// ---------------- pass 2: per-pair attention algebra, one wave per pair ----
// scores = Wq*G*Wk^T + (Wq s)bk^T + bq(Wk s)^T + HW*bq*bk^T ; softmax rows
// M = attn*Wv, d = attn*bv ; emit diag(8) + symmetrized offdiag(28) + d(8)
__global__ void __launch_bounds__(256) lcsap_pass2(
    const float* __restrict__ wsG, const float* __restrict__ Wq,
    const float* __restrict__ bq, const float* __restrict__ Wk,
    const float* __restrict__ bk, const float* __restrict__ Wv,
    const float* __restrict__ bv, float* __restrict__ wsM) {
  __shared__ float stage[8][144];  // per wave: [0..63] tile, [64..71] u,
                                   // [72..79] w, [80..143] M
  const int lane = threadIdx.x & 31;
  const int lw = threadIdx.x >> 5;
  const int pair = blockIdx.x * 8 + lw;   // grid = 64 blocks -> 512 waves
  const int n = pair & (NC - 1);
  const float* Gt = wsG + (size_t)pair * PRM;
  const float* sv = Gt + 36;
  const float* wq = Wq + n * 64;
  const float* wk = Wk + n * 64;
  const float* wv = Wv + n * 64;
  float* st = stage[lw];

  const int row = lane & 15;   // A row / D row group
  const int col = lane & 15;   // B col / D col
  const int hf = lane >> 4;    // K half selector within a 4-K step
  const bool lo8 = (lane < 8);

  // ---- P = Wq(8x8) * G(8x8) via 2 chained 16x16x4 WMMAs ----
  v8f acc1 = {};
#pragma unroll
  for (int kk = 0; kk < 8; kk += 4) {
    const int k0 = kk + hf * 2;
    v2f a, bb;
    a.x = (row < 8) ? wq[row * 8 + k0] : 0.f;
    a.y = (row < 8) ? wq[row * 8 + k0 + 1] : 0.f;
    bb.x = (col < 8) ? getG(Gt, k0, col) : 0.f;
    bb.y = (col < 8) ? getG(Gt, k0 + 1, col) : 0.f;
    acc1 = wmma4(a, bb, acc1);
  }
  if (lo8) {  // stage P rows 0..7 (lanes 0..7 hold cols 0..7)
#pragma unroll
    for (int i = 0; i < 8; ++i) st[i * 8 + lane] = acc1[i];
    float u = 0.f, w = 0.f;   // u = Wq*s, w = Wk*s
#pragma unroll
    for (int c = 0; c < 8; ++c) {
      u = fmaf(wq[lane * 8 + c], sv[c], u);
      w = fmaf(wk[lane * 8 + c], sv[c], w);
    }
    st[64 + lane] = u;
    st[72 + lane] = w;
  }
  __syncthreads();

  // ---- S0 = P * Wk^T ----
  v8f acc2 = {};
#pragma unroll
  for (int kk = 0; kk < 8; kk += 4) {
    const int k0 = kk + hf * 2;
    v2f a, bb;
    a.x = (row < 8) ? st[row * 8 + k0] : 0.f;
    a.y = (row < 8) ? st[row * 8 + k0 + 1] : 0.f;
    bb.x = (col < 8) ? wk[col * 8 + k0] : 0.f;
    bb.y = (col < 8) ? wk[col * 8 + k0 + 1] : 0.f;
    acc2 = wmma4(a, bb, acc2);
  }
  __syncthreads();

  // ---- full scores + row softmax; attn rows -> st[0..63] ----
  const float bkj = bk[n * 8 + (lane & 7)];
  const float wj = st[72 + (lane & 7)];
#pragma unroll
  for (int i = 0; i < 8; ++i) {
    const float ui = st[64 + i];
    const float bqi = bq[n * 8 + i];
    float sc = acc2[i] + ui * bkj + bqi * wj + 9216.0f * bqi * bkj;
    sc *= (1.0f / 96.0f);                 // 1/sqrt(96*96)
    float mv = lo8 ? sc : -3.0e38f;
#pragma unroll
    for (int m = 4; m >= 1; m >>= 1) mv = fmaxf(mv, __shfl_xor(mv, m, 32));
    const float e = lo8 ? __expf(sc - mv) : 0.f;
    float es = e;
#pragma unroll
    for (int m = 4; m >= 1; m >>= 1) es += __shfl_xor(es, m, 32);
    if (lo8) st[i * 8 + lane] = e / es;
  }
  __syncthreads();

  // ---- M = attn * Wv ----
  v8f acc3 = {};
#pragma unroll
  for (int kk = 0; kk < 8; kk += 4) {
    const int k0 = kk + hf * 2;
    v2f a, bb;
    a.x = (row < 8) ? st[row * 8 + k0] : 0.f;
    a.y = (row < 8) ? st[row * 8 + k0 + 1] : 0.f;
    bb.x = (col < 8) ? wv[k0 * 8 + col] : 0.f;
    bb.y = (col < 8) ? wv[(k0 + 1) * 8 + col] : 0.f;
    acc3 = wmma4(a, bb, acc3);
  }

  float* prm = wsM + (size_t)pair * PRM;
  if (lo8) {
#pragma unroll
    for (int i = 0; i < 8; ++i) st[80 + i * 8 + lane] = acc3[i];  // stage M
    float dq = 0.f;   // d = attn * bv  (attn still in st[0..63])
#pragma unroll
    for (int k = 0; k < 8; ++k) dq = fmaf(st[lane * 8 + k], bv[n * 8 + k], dq);
    prm[36 + lane] = dq;
  }
  __syncthreads();
  if (lo8) prm[lane] = st[80 + lane * 9];  // diag(M)
  if (lane < 28) {                         // offdiag: M[i][j] + M[j][i]
    int t = lane, i = 0;
    while (t >= 7 - i) { t -= 7 - i; ++i; }
    const int j = i + 1 + t;
    prm[8 + lane] = st[80 + i * 8 + j] + st[80 + j * 8 + i];
  }
}

// ---------------- pass 3: pooled[p] = x_p^T M x_p + d.x_p (streaming) ------
__global__ void __launch_bounds__(256) lcsap_pass3(const float* __restrict__ x,
                                                   const float* __restrict__ wsM,
                                                   float* __restrict__ out) {
  const int blk = blockIdx.x;
  const int pair = blk / SPLIT3;
  const int slice = blk % SPLIT3;
  const int b = pair >> 6, n = pair & 63;
  const float* prm = wsM + (size_t)pair * PRM;

  float diag[8], off[28], dv[8];
#pragma unroll
  for (int i = 0; i < 8; ++i) diag[i] = prm[i];
#pragma unroll
  for (int t = 0; t < 28; ++t) off[t] = prm[8 + t];
#pragma unroll
  for (int i = 0; i < 8; ++i) dv[i] = prm[36 + i];

  const float* xb = x + (size_t)(b * TDIM + n * CS) * HWP;
  float* ob = out + (size_t)pair * HWP;
  const int p0 = slice * 1024 + threadIdx.x * 4;

  float v[8][4];
#pragma unroll
  for (int c = 0; c < 8; ++c) {
    float4 t4 = *reinterpret_cast<const float4*>(xb + c * HWP + p0);
    v[c][0] = t4.x; v[c][1] = t4.y; v[c][2] = t4.z; v[c][3] = t4.w;
  }
  float rr[4];
#pragma unroll
  for (int px = 0; px < 4; ++px) {
    float acc = 0.f;
    int t = 0;
#pragma unroll
    for (int i = 0; i < 8; ++i) {
      float z = fmaf(diag[i], v[i][px], dv[i]);
#pragma unroll
      for (int j = i + 1; j < 8; ++j) { z = fmaf(off[t], v[j][px], z); ++t; }
      acc = fmaf(v[i][px], z, acc);
    }
    rr[px] = acc;
  }
  float4 r;
  r.x = rr[0]; r.y = rr[1]; r.z = rr[2]; r.w = rr[3];
  *reinterpret_cast<float4*>(ob + p0) = r;
}

// ---------------- host entry ----------------
extern "C" void kernel_launch(void* const* d_in, const int* in_sizes, int n_in,
                              void* d_out, int out_size, void* d_ws,
                              size_t ws_size, hipStream_t stream) {
  const float* x  = (const float*)d_in[0];
  const float* Wq = (const float*)d_in[1];
  const float* bq = (const float*)d_in[2];
  const float* Wk = (const float*)d_in[3];
  const float* bk = (const float*)d_in[4];
  const float* Wv = (const float*)d_in[5];
  const float* bv = (const float*)d_in[6];
  float* out = (float*)d_out;

  float* wsG = (float*)d_ws;                 // 512*44 floats
  float* wsM = wsG + (size_t)NPAIR * PRM;    // 512*44 floats

  lcsap_pass1<<<NPAIR, 256, 0, stream>>>(x, wsG);
  lcsap_pass2<<<NPAIR / 8, 256, 0, stream>>>(wsG, Wq, bq, Wk, bk, Wv, bv, wsM);
  lcsap_pass3<<<NPAIR * SPLIT3, 256, 0, stream>>>(x, wsM, out);
}